// S4_layer_38190849196291
// MI455X (gfx1250) — compile-verified
//
#include <hip/hip_runtime.h>

// ---------------------------------------------------------------------------
// S4 layer on MI455X (gfx1250), fp32 WMMA 16x16x4 everywhere.
//   bx = x @ B^T            (WMMA GEMM, 65536x64x256)
//   h-recurrence            (blocked parallel scan, T=32, NC=64 chunks,
//                            u-tiles software-pipelined into registers)
//   y  = hs @ C^T           (WMMA GEMM, 65536x256x64)
// ---------------------------------------------------------------------------

typedef __attribute__((ext_vector_type(2))) float v2f;
typedef __attribute__((ext_vector_type(8))) float v8f;

constexpr int S    = 2048;
constexpr int BT   = 32;     // batch
constexpr int DIN  = 256;
constexpr int DOUT = 256;
constexpr int LAT  = 64;     // latent
constexpr int T    = 32;     // chunk length
constexpr int NC   = 64;     // number of chunks (T*NC == S)
constexpr int HPAD = 68;     // LDS row stride for h (avoid bank conflicts)

// ---- WMMA f32 16x16x4 wrapper -------------------------------------------
__device__ __forceinline__ v8f wmma4(v2f a, v2f b, v8f c) {
  // (neg_a, A, neg_b, B, c_mod, C, reuse_a, reuse_b)
  return __builtin_amdgcn_wmma_f32_16x16x4_f32(false, a, false, b, (short)0, c,
                                               false, false);
}

// A-frag from LDS h buffer (rows mt*16..+15, stride HPAD), 16 K-steps of 4
__device__ __forceinline__ v8f tile_mm_lds(const float* hbase, const v2f* bfrag,
                                           v8f acc, int mt, int m, int hi) {
  const float* ap = hbase + (mt * 16 + m) * HPAD + 2 * hi;
#pragma unroll
  for (int kk = 0; kk < 16; ++kk) {
    v2f a = *(const v2f*)(ap + kk * 4);
    acc = wmma4(a, bfrag[kk], acc);
  }
  return acc;
}

// Preload 16 B-fragments of a 64x64 weight W (row-major, W[n][k]) for tile nt
__device__ __forceinline__ void load_bfrag64(v2f* bfrag, const float* W, int nt,
                                             int m, int hi) {
#pragma unroll
  for (int kk = 0; kk < 16; ++kk)
    bfrag[kk] = *(const v2f*)(W + (nt * 16 + m) * 64 + kk * 4 + 2 * hi);
}

// Load a 16x16 tile of a [32][64] row-major matrix into the C accumulator
__device__ __forceinline__ v8f load_u_tile(const float* u, int mt, int nt,
                                           int m, int hi) {
  v8f acc;
#pragma unroll
  for (int r = 0; r < 8; ++r)
    acc[r] = u[(mt * 16 + r + 8 * hi) * 64 + nt * 16 + m];
  return acc;
}

__device__ __forceinline__ void store_tile_lds(float* hbase, v8f d, int mt,
                                               int nt, int m, int hi) {
#pragma unroll
  for (int r = 0; r < 8; ++r)
    hbase[(mt * 16 + r + 8 * hi) * HPAD + nt * 16 + m] = d[r];
}

__device__ __forceinline__ void store_tile_g64(float* g, v8f d, int mt, int nt,
                                               int m, int hi) {
#pragma unroll
  for (int r = 0; r < 8; ++r)
    g[(mt * 16 + r + 8 * hi) * 64 + nt * 16 + m] = d[r];
}

// ---------------------------------------------------------------------------
// Kernel 1: bx[s*BT + b][l] = sum_d x[s*BT+b][d] * Bw[l][d]
// one wave per 16x16 output tile; M=65536 (mt 0..4095), N=64 (nt 0..3)
// ---------------------------------------------------------------------------
__global__ void bx_gemm_k(const float* __restrict__ x,
                          const float* __restrict__ Bw,
                          float* __restrict__ bx) {
  int tid = threadIdx.x;
  int lane = tid & 31, hi = lane >> 4, m = lane & 15;
  int w = (blockIdx.x * blockDim.x + tid) >> 5;
  int mt = w >> 2, nt = w & 3;

  const float* ap = x + (size_t)(mt * 16 + m) * DIN + 2 * hi;
  const float* bp = Bw + (size_t)(nt * 16 + m) * DIN + 2 * hi;

  v8f acc = {0.f, 0.f, 0.f, 0.f, 0.f, 0.f, 0.f, 0.f};
#pragma unroll 8
  for (int kk = 0; kk < DIN / 4; ++kk) {
    v2f a = *(const v2f*)(ap + kk * 4);
    v2f b = *(const v2f*)(bp + kk * 4);
    acc = wmma4(a, b, acc);
  }
  store_tile_g64(bx + (size_t)mt * 16 * LAT, acc, 0, nt, m, hi);
}

// ---------------------------------------------------------------------------
// Kernel 2: Apow = A^T  (T = 32 = 2^5, five LDS squarings). Tiny; VALU.
// Convention: step is h_new[m][n] = sum_k h[m][k]*W[n][k]; T applications of
// W=A equal one application of W=A^T (standard matrix power of A).
// ---------------------------------------------------------------------------
__global__ void matpow_k(const float* __restrict__ A, float* __restrict__ Apow) {
  __shared__ float P[2][64][65];
  int tid = threadIdx.x;
  for (int i = tid; i < 64 * 64; i += blockDim.x)
    P[0][i >> 6][i & 63] = A[i];
  __syncthreads();
  int p = 0;
#pragma unroll 1
  for (int s = 0; s < 5; ++s) {
    for (int i = tid; i < 64 * 64; i += blockDim.x) {
      int r = i >> 6, c = i & 63;
      float acc = 0.f;
#pragma unroll 8
      for (int k = 0; k < 64; ++k) acc += P[p][r][k] * P[p][k][c];
      P[1 - p][r][c] = acc;
    }
    __syncthreads();
    p ^= 1;
  }
  for (int i = tid; i < 64 * 64; i += blockDim.x)
    Apow[i] = P[p][i >> 6][i & 63];
}

// ---------------------------------------------------------------------------
// Kernel 3: per-chunk local pass from h=0; emits L_c = state after T steps.
// 8 waves: (mt,nt) = (w>>2, w&3). A-weights resident in 16 v2f regs per wave.
// u[t+1] tile prefetched into registers while step t computes.
// ---------------------------------------------------------------------------
__global__ void chunk_local_k(const float* __restrict__ bx,
                              const float* __restrict__ A,
                              float* __restrict__ Lc) {
  __shared__ float hbuf[2][BT * HPAD];
  int tid = threadIdx.x;
  int lane = tid & 31, hi = lane >> 4, m = lane & 15;
  int w = tid >> 5, mt = w >> 2, nt = w & 3;
  int c = blockIdx.x;

  for (int i = tid; i < BT * HPAD; i += blockDim.x) hbuf[0][i] = 0.f;

  v2f bfrag[16];
  load_bfrag64(bfrag, A, nt, m, hi);

  v8f u_next = load_u_tile(bx + (size_t)(c * T) * BT * LAT, mt, nt, m, hi);
  __syncthreads();

  int p = 0;
#pragma unroll 1
  for (int i = 0; i < T; ++i) {
    v8f acc = u_next;
    if (i + 1 < T)  // issue next step's global loads before this step's math
      u_next =
          load_u_tile(bx + (size_t)(c * T + i + 1) * BT * LAT, mt, nt, m, hi);
    acc = tile_mm_lds(hbuf[p], bfrag, acc, mt, m, hi);
    store_tile_lds(hbuf[1 - p], acc, mt, nt, m, hi);
    __syncthreads();
    p ^= 1;
  }
  float* out = Lc + (size_t)c * BT * LAT;
  for (int i = tid; i < BT * LAT; i += blockDim.x)
    out[i] = hbuf[p][(i >> 6) * HPAD + (i & 63)];
}

// ---------------------------------------------------------------------------
// Kernel 4: sequential scan over chunk boundaries (NC-1 dependent steps):
//   carries[0] = 0;  carries[c+1] = carries[c] @ Apow^T + L_c
// ---------------------------------------------------------------------------
__global__ void chunk_scan_k(const float* __restrict__ Apow,
                             const float* __restrict__ Lc,
                             float* __restrict__ carries) {
  __shared__ float hbuf[2][BT * HPAD];
  int tid = threadIdx.x;
  int lane = tid & 31, hi = lane >> 4, m = lane & 15;
  int w = tid >> 5, mt = w >> 2, nt = w & 3;

  for (int i = tid; i < BT * HPAD; i += blockDim.x) hbuf[0][i] = 0.f;
  for (int i = tid; i < BT * LAT; i += blockDim.x) carries[i] = 0.f;

  v2f bfrag[16];
  load_bfrag64(bfrag, Apow, nt, m, hi);

  v8f u_next = load_u_tile(Lc, mt, nt, m, hi);
  __syncthreads();

  int p = 0;
#pragma unroll 1
  for (int c = 0; c < NC - 1; ++c) {
    v8f acc = u_next;
    if (c + 1 < NC - 1)
      u_next = load_u_tile(Lc + (size_t)(c + 1) * BT * LAT, mt, nt, m, hi);
    acc = tile_mm_lds(hbuf[p], bfrag, acc, mt, m, hi);
    store_tile_lds(hbuf[1 - p], acc, mt, nt, m, hi);
    store_tile_g64(carries + (size_t)(c + 1) * BT * LAT, acc, mt, nt, m, hi);
    __syncthreads();
    p ^= 1;
  }
}

// ---------------------------------------------------------------------------
// Kernel 5: per-chunk final pass from exact carry; emits all hs[t] = h[t+1]
// into the h-region of d_out (layout [S+1][BT][LAT], slot 0 pre-zeroed).
// ---------------------------------------------------------------------------
__global__ void chunk_final_k(const float* __restrict__ bx,
                              const float* __restrict__ A,
                              const float* __restrict__ carries,
                              float* __restrict__ h_out) {
  __shared__ float hbuf[2][BT * HPAD];
  int tid = threadIdx.x;
  int lane = tid & 31, hi = lane >> 4, m = lane & 15;
  int w = tid >> 5, mt = w >> 2, nt = w & 3;
  int c = blockIdx.x;

  const float* cin = carries + (size_t)c * BT * LAT;
  for (int i = tid; i < BT * LAT; i += blockDim.x)
    hbuf[0][(i >> 6) * HPAD + (i & 63)] = cin[i];

  v2f bfrag[16];
  load_bfrag64(bfrag, A, nt, m, hi);

  v8f u_next = load_u_tile(bx + (size_t)(c * T) * BT * LAT, mt, nt, m, hi);
  __syncthreads();

  int p = 0;
#pragma unroll 1
  for (int i = 0; i < T; ++i) {
    int t = c * T + i;
    v8f acc = u_next;
    if (i + 1 < T)
      u_next = load_u_tile(bx + (size_t)(t + 1) * BT * LAT, mt, nt, m, hi);
    acc = tile_mm_lds(hbuf[p], bfrag, acc, mt, m, hi);
    store_tile_lds(hbuf[1 - p], acc, mt, nt, m, hi);
    store_tile_g64(h_out + (size_t)(t + 1) * BT * LAT, acc, mt, nt, m, hi);
    __syncthreads();
    p ^= 1;
  }
}

// ---------------------------------------------------------------------------
// Kernel 6: y[r][o] = sum_l hs[r][l] * C[o][l];  M=65536 (mt), N=256 (nt 0..15)
// ---------------------------------------------------------------------------
__global__ void y_gemm_k(const float* __restrict__ hs,
                         const float* __restrict__ C,
                         float* __restrict__ y) {
  int tid = threadIdx.x;
  int lane = tid & 31, hi = lane >> 4, m = lane & 15;
  int w = (blockIdx.x * blockDim.x + tid) >> 5;
  int mt = w >> 4, nt = w & 15;

  const float* ap = hs + (size_t)(mt * 16 + m) * LAT + 2 * hi;
  const float* bp = C + (size_t)(nt * 16 + m) * LAT + 2 * hi;

  v8f acc = {0.f, 0.f, 0.f, 0.f, 0.f, 0.f, 0.f, 0.f};
#pragma unroll
  for (int kk = 0; kk < LAT / 4; ++kk) {
    v2f a = *(const v2f*)(ap + kk * 4);
    v2f b = *(const v2f*)(bp + kk * 4);
    acc = wmma4(a, b, acc);
  }
#pragma unroll
  for (int r = 0; r < 8; ++r)
    y[(size_t)(mt * 16 + r + 8 * hi) * DOUT + nt * 16 + m] = acc[r];
}

// ---------------------------------------------------------------------------
extern "C" void kernel_launch(void* const* d_in, const int* in_sizes, int n_in,
                              void* d_out, int out_size, void* d_ws,
                              size_t ws_size, hipStream_t stream) {
  const float* x  = (const float*)d_in[0];
  const float* A  = (const float*)d_in[1];
  const float* Bw = (const float*)d_in[2];
  const float* C  = (const float*)d_in[3];

  float* out      = (float*)d_out;
  float* y_region = out;                           // [S][BT][DOUT]
  float* h_region = out + (size_t)S * BT * DOUT;   // [S+1][BT][LAT]

  // bx (16 MB) staged in the y region (64 MB) — dead before y is written.
  float* bx = y_region;

  float* ws      = (float*)d_ws;
  float* Apow    = ws;                                   // 64*64
  float* Lc      = ws + 64 * 64;                         // NC*BT*LAT
  float* carries = ws + 64 * 64 + (size_t)NC * BT * LAT; // NC*BT*LAT

  // h[0] = 0
  hipMemsetAsync(h_region, 0, (size_t)BT * LAT * sizeof(float), stream);

  bx_gemm_k<<<(S * BT / 16) * (LAT / 16) / 8, 256, 0, stream>>>(x, Bw, bx);
  matpow_k<<<1, 256, 0, stream>>>(A, Apow);
  chunk_local_k<<<NC, 256, 0, stream>>>(bx, A, Lc);
  chunk_scan_k<<<1, 256, 0, stream>>>(Apow, Lc, carries);
  chunk_final_k<<<NC, 256, 0, stream>>>(bx, A, carries, h_region);
  y_gemm_k<<<(S * BT / 16) * (DOUT / 16) / 8, 256, 0, stream>>>(
      h_region + BT * LAT, C, y_region);
}